// ParticleCollision_83227876262109
// MI455X (gfx1250) — compile-verified
//
#include <hip/hip_runtime.h>
#include <stdint.h>

#define NPART 4096
#define NBATCH 2
#define MAXC 128
#define RADIUS 0.1f

typedef float v2f __attribute__((ext_vector_type(2)));
typedef float v8f __attribute__((ext_vector_type(8)));

// ---------------------------------------------------------------------------
// Kernel 1: per-batch bounds reduction + cell ids + stable sort keys
// ---------------------------------------------------------------------------
__global__ void pc_bounds_cells(const float* __restrict__ locs,
                                uint32_t* __restrict__ keys) {
  const int b   = blockIdx.x;
  const int tid = threadIdx.x;
  const float* L = locs + (size_t)b * NPART * 3;

  __shared__ float smn[3][256];
  __shared__ float smx[3][256];

  float mn[3] = { 1e30f,  1e30f,  1e30f};
  float mx[3] = {-1e30f, -1e30f, -1e30f};
  for (int p = tid; p < NPART; p += 256) {
#pragma unroll
    for (int d = 0; d < 3; ++d) {
      float v = L[p * 3 + d];
      mn[d] = fminf(mn[d], v);
      mx[d] = fmaxf(mx[d], v);
    }
  }
#pragma unroll
  for (int d = 0; d < 3; ++d) { smn[d][tid] = mn[d]; smx[d][tid] = mx[d]; }
  __syncthreads();
  for (int s = 128; s > 0; s >>= 1) {
    if (tid < s) {
#pragma unroll
      for (int d = 0; d < 3; ++d) {
        smn[d][tid] = fminf(smn[d][tid], smn[d][tid + s]);
        smx[d][tid] = fmaxf(smx[d][tid], smx[d][tid + s]);
      }
    }
    __syncthreads();
  }

  __shared__ float lo2[3];
  __shared__ float gdf[3];
  __shared__ int   str[3];
  if (tid == 0) {
    float gd[3];
#pragma unroll
    for (int d = 0; d < 3; ++d) {
      float l = smn[d][0], u = smx[d][0];
      float gdim = ceilf(fminf(fmaxf((u - l) / RADIUS, 0.0f), 96.0f));
      float c = (l + u) * 0.5f;
      lo2[d] = c - gdim * RADIUS * 0.5f;   // uses grid_dims BEFORE max(.,1)
      gd[d]  = fmaxf(gdim, 1.0f);
      gdf[d] = gd[d];
    }
    int g1 = (int)gd[1], g2 = (int)gd[2];
    str[0] = g1 * g2;  // row-major strides
    str[1] = g2;
    str[2] = 1;
  }
  __syncthreads();

  for (int p = tid; p < NPART; p += 256) {
    int id = 0;
#pragma unroll
    for (int d = 0; d < 3; ++d) {
      float cc = floorf((L[p * 3 + d] - lo2[d]) / RADIUS);
      cc = fminf(fmaxf(cc, 0.0f), gdf[d] - 1.0f);
      id += (int)cc * str[d];
    }
    // key = cell_id * 4096 + original index  -> unique, stable-by-construction
    keys[(size_t)b * NPART + p] = (uint32_t)id * 4096u + (uint32_t)p;
  }
}

// ---------------------------------------------------------------------------
// Kernel 2: single-block bitonic sort per batch + gathers
// ---------------------------------------------------------------------------
__global__ void pc_sort_gather(const float* __restrict__ locs,
                               const float* __restrict__ data,
                               const uint32_t* __restrict__ keys,
                               float* __restrict__ out_idxs,
                               float* __restrict__ out_locs,
                               float* __restrict__ out_data,
                               float4* __restrict__ ws_locs4,
                               float* __restrict__ ws_sq) {
  const int b   = blockIdx.x;
  const int tid = threadIdx.x;
  __shared__ uint32_t sk[NPART];

  for (int i = tid; i < NPART; i += 1024) sk[i] = keys[(size_t)b * NPART + i];
  __syncthreads();

  for (unsigned k = 2; k <= NPART; k <<= 1) {
    for (unsigned j = k >> 1; j > 0; j >>= 1) {
      for (unsigned i = tid; i < NPART; i += 1024) {
        unsigned p = i ^ j;
        if (p > i) {
          uint32_t x = sk[i], y = sk[p];
          bool asc = ((i & k) == 0);
          if ((x > y) == asc) { sk[i] = y; sk[p] = x; }
        }
      }
      __syncthreads();
    }
  }

  for (int i = tid; i < NPART; i += 1024) {
    uint32_t ord = sk[i] & 4095u;
    size_t src = (size_t)b * NPART + ord;
    size_t dst = (size_t)b * NPART + i;
    out_idxs[dst] = (float)ord;
    float x = locs[src * 3 + 0];
    float y = locs[src * 3 + 1];
    float z = locs[src * 3 + 2];
    out_locs[dst * 3 + 0] = x;
    out_locs[dst * 3 + 1] = y;
    out_locs[dst * 3 + 2] = z;
    ws_locs4[dst] = make_float4(x, y, z, 0.0f);
    ws_sq[dst]    = x * x + y * y + z * z;
    const float4* dsrc = (const float4*)(data + src * 16);
    float4*       ddst = (float4*)(out_data + dst * 16);
#pragma unroll
    for (int q = 0; q < 4; ++q) ddst[q] = dsrc[q];
  }
}

// ---------------------------------------------------------------------------
// Async direct-to-LDS helper (gfx1250): VDST = 32-bit LDS byte offset,
// VADDR = 64-bit global address, GV mode (no SADDR). Tracked by ASYNCcnt.
// ---------------------------------------------------------------------------
__device__ __forceinline__ void async_copy_b128(void* lds_dst, const void* gsrc) {
  uint32_t l = (uint32_t)(uintptr_t)lds_dst;   // low 32 bits = LDS offset
  uint64_t g = (uint64_t)(uintptr_t)gsrc;
  asm volatile("global_load_async_to_lds_b128 %0, %1, off"
               :: "v"(l), "v"(g) : "memory");
}

__device__ __forceinline__ void async_wait_all() {
  asm volatile("s_wait_asynccnt 0" ::: "memory");
}

// ---------------------------------------------------------------------------
// Kernel 3: WMMA Gram tiles (f32 16x16x4) + first-128 neighbor collection
// 256 threads = 8 waves; each wave owns a 16-row tile, loops 256 column tiles.
// LDS fill uses GLOBAL_LOAD_ASYNC_TO_LDS_B128 (no VGPR round-trip).
// ---------------------------------------------------------------------------
__global__ void pc_neighbors(const float4* __restrict__ locs4,
                             const float* __restrict__ sqv,
                             float* __restrict__ out_nbr) {
  extern __shared__ char smem[];
  float4* tile = (float4*)smem;
  float*  ssq  = (float*)(smem + NPART * sizeof(float4));

  const int b   = blockIdx.x >> 5;   // 32 blocks per batch
  const int blk = blockIdx.x & 31;
  const size_t base = (size_t)b * NPART;

  // Stage 64KB of float4 positions + 16KB of |r|^2 into LDS asynchronously.
  for (int i = threadIdx.x; i < NPART; i += 256)
    async_copy_b128(tile + i, locs4 + base + i);
  for (int i = threadIdx.x; i < NPART / 4; i += 256)
    async_copy_b128(ssq + i * 4, sqv + base + i * 4);
  async_wait_all();
  __syncthreads();

  const int wave = threadIdx.x >> 5;
  const int lane = threadIdx.x & 31;
  const int li   = lane & 15;
  const int hi   = lane >> 4;        // 0: lanes 0-15 (K0,K1), 1: lanes 16-31 (K2,K3)
  const int rbase = (blk * 8 + wave) * 16;

  // A matrix 16x4 f32: lane M = lane&15; VGPR0/1 hold K pair (x,y) or (z,0)
  float4 pr = tile[rbase + li];
  v2f a = { hi ? pr.z : pr.x, hi ? 0.0f : pr.y };

  // Row |r|^2 for C/D layout rows: VGPR v <-> row v + 8*hi
  float srow[8];
#pragma unroll
  for (int v = 0; v < 8; ++v) srow[v] = ssq[rbase + v + 8 * hi];

  int cnt = (lane < 16) ? 0 : MAXC;  // lanes 16-31 report "full"
  float* nrow = out_nbr + (((size_t)b * NPART) + rbase + li) * MAXC;

  const float R2 = RADIUS * RADIUS;

  for (int ct = 0; ct < NPART / 16; ++ct) {
    const int cbase = ct * 16;
    float4 pc = tile[cbase + li];
    v2f bm = { hi ? pc.z : pc.x, hi ? 0.0f : pc.y };
    float sqc = ssq[cbase + li];

    v8f acc = {0.f, 0.f, 0.f, 0.f, 0.f, 0.f, 0.f, 0.f};
    // D = A(16x4) * B(4x16) + 0  -> 16x16 dot-product tile
    v8f g = __builtin_amdgcn_wmma_f32_16x16x4_f32(
        false, a, false, bm, (short)0, acc, false, false);

    unsigned bal0, bal1, bal2, bal3, bal4, bal5, bal6, bal7;
    {
      float d2;
      d2 = srow[0] + sqc - 2.0f * g[0]; bal0 = (unsigned)__ballot(d2 <= R2);
      d2 = srow[1] + sqc - 2.0f * g[1]; bal1 = (unsigned)__ballot(d2 <= R2);
      d2 = srow[2] + sqc - 2.0f * g[2]; bal2 = (unsigned)__ballot(d2 <= R2);
      d2 = srow[3] + sqc - 2.0f * g[3]; bal3 = (unsigned)__ballot(d2 <= R2);
      d2 = srow[4] + sqc - 2.0f * g[4]; bal4 = (unsigned)__ballot(d2 <= R2);
      d2 = srow[5] + sqc - 2.0f * g[5]; bal5 = (unsigned)__ballot(d2 <= R2);
      d2 = srow[6] + sqc - 2.0f * g[6]; bal6 = (unsigned)__ballot(d2 <= R2);
      d2 = srow[7] + sqc - 2.0f * g[7]; bal7 = (unsigned)__ballot(d2 <= R2);
    }

    if (lane < 16) {
      // lane L owns row rbase+L; VGPR v maps to rows v (bits 0-15) and v+8 (bits 16-31)
      unsigned r = (unsigned)(li & 7);
      unsigned bb = bal0;
      bb = (r == 1u) ? bal1 : bb;
      bb = (r == 2u) ? bal2 : bb;
      bb = (r == 3u) ? bal3 : bb;
      bb = (r == 4u) ? bal4 : bb;
      bb = (r == 5u) ? bal5 : bb;
      bb = (r == 6u) ? bal6 : bb;
      bb = (r == 7u) ? bal7 : bb;
      unsigned m = (li < 8) ? (bb & 0xFFFFu) : (bb >> 16);
      while (m && cnt < MAXC) {
        int c = __builtin_ctz(m);
        nrow[cnt++] = (float)(cbase + c);
        m &= m - 1u;
      }
    }
    // wave-uniform early exit keeps EXEC all-ones at the WMMA
    if ((unsigned)__ballot(cnt < MAXC) == 0u) break;
  }

  if (lane < 16) {
    for (int k = cnt; k < MAXC; ++k) nrow[k] = -1.0f;
  }
}

// ---------------------------------------------------------------------------
extern "C" void kernel_launch(void* const* d_in, const int* in_sizes, int n_in,
                              void* d_out, int out_size, void* d_ws, size_t ws_size,
                              hipStream_t stream) {
  (void)in_sizes; (void)n_in; (void)out_size; (void)ws_size;

  const float* locs = (const float*)d_in[0];   // [2,4096,3]
  const float* data = (const float*)d_in[1];   // [2,4096,16]

  float* out      = (float*)d_out;
  float* out_idxs = out;                                        // 8192
  float* out_nbr  = out + (size_t)NBATCH * NPART;               // 1,048,576
  float* out_locs = out_nbr + (size_t)NBATCH * NPART * MAXC;    // 24,576
  float* out_data = out_locs + (size_t)NBATCH * NPART * 3;      // 131,072

  uint8_t*  w        = (uint8_t*)d_ws;
  uint32_t* keys     = (uint32_t*)w;                             // 32 KB
  float4*   ws_locs4 = (float4*)(w + 32768);                     // 128 KB
  float*    ws_sq    = (float*)(w + 32768 + 131072);             // 32 KB

  pc_bounds_cells<<<NBATCH, 256, 0, stream>>>(locs, keys);
  pc_sort_gather<<<NBATCH, 1024, 0, stream>>>(locs, data, keys, out_idxs,
                                              out_locs, out_data, ws_locs4, ws_sq);
  // 8 waves/block, 16 rows/wave -> 32 blocks per batch
  pc_neighbors<<<NBATCH * 32, 256, NPART * sizeof(float4) + NPART * sizeof(float),
                 stream>>>(ws_locs4, ws_sq, out_nbr);
}